// Regressor_86431921864892
// MI455X (gfx1250) — compile-verified
//
#include <hip/hip_runtime.h>
#include <hip/hip_bf16.h>
#include <math.h>

typedef __attribute__((ext_vector_type(2))) float v2f;
typedef __attribute__((ext_vector_type(8))) float v8f;

#define NJ      21
#define PF      512
#define JC      515            // real cols per joint (512 feat + 3 img)
#define JCP     516            // padded to multiple of 4
#define KPAD    (NJ * JCP)     // 10836
#define FEATD   (NJ * JC)      // 10815
#define DPAD    (FEATD + 18)   // 10833
#define NREG    96             // 16 joints * 6 outputs
#define BATCH   1024
#define MTOT    (3 * BATCH)    // 3072
#define KCHUNKS 3
#define JPC     7              // joints per K-chunk

// ---- output layout (flat, reference tuple order) ----
#define SH_OFF   0                       // 3 x [1024,10]
#define POSE_OFF 30720                   // 3 x [1024,48]
#define CAM_OFF  178176                  // 3 x [1024,3]

// ---- workspace layout (bytes) ----
#define WS_POOLED 0                       // 1024*2048 f32 = 8 MB
#define WS_WP     (8u * 1024u * 1024u)    // KPAD*96 f32 = 4.16 MB
#define WS_G      (WS_WP + 4200448u)      // 3 x 3072*96 f32 = 3.54 MB

// ------------------------------------------------------------------
// Kernel 1: mean pool over 8x8  (bandwidth bound: 536 MB)
// ------------------------------------------------------------------
__global__ __launch_bounds__(256)
void pool_kernel(const float* __restrict__ x, float* __restrict__ pooled) {
    int idx = blockIdx.x * 256 + threadIdx.x;          // b*2048 + c
    if (idx >= BATCH * 2048) return;
    const float4* p = (const float4*)(x + (size_t)idx * 64);
    float s = 0.0f;
#pragma unroll
    for (int i = 0; i < 16; ++i) { float4 v = p[i]; s += v.x + v.y + v.z + v.w; }
    pooled[idx] = s * (1.0f / 64.0f);
}

// ------------------------------------------------------------------
// Kernel 2: shape (30) + cam (9) heads from pooled row (block per b)
// ------------------------------------------------------------------
__global__ __launch_bounds__(256)
void shapecam_kernel(const float* __restrict__ pooled,
                     const float* __restrict__ shape_w, const float* __restrict__ shape_b,
                     const float* __restrict__ cam_w,   const float* __restrict__ cam_b,
                     float* __restrict__ out) {
    __shared__ float row[2048];
    int b = blockIdx.x;
    for (int i = threadIdx.x; i < 2048; i += 256) row[i] = pooled[(size_t)b * 2048 + i];
    __syncthreads();
    int wave = threadIdx.x >> 5;
    int lane = threadIdx.x & 31;
    for (int o = wave; o < 39; o += 8) {
        const float* w = (o < 30) ? (shape_w + (size_t)o * 2048)
                                  : (cam_w + (size_t)(o - 30) * 2048);
        float s = 0.0f;
        for (int i = lane; i < 2048; i += 32) s += row[i] * w[i];
#pragma unroll
        for (int off = 16; off > 0; off >>= 1) s += __shfl_down(s, off, 32);
        if (lane == 0) {
            if (o < 30) {
                int st = o / 10, oo = o % 10;
                out[SH_OFF + st * (BATCH * 10) + b * 10 + oo] = s + shape_b[o];
            } else {
                int oc = o - 30; int st = oc / 3, oo = oc % 3;
                out[CAM_OFF + st * (BATCH * 3) + b * 3 + oo] = s + cam_b[oc];
            }
        }
    }
}

// ------------------------------------------------------------------
// Kernel 3: pack regressor weights, K-pair interleaved so each lane's
// B fragment (K=k,k+1 at column n) is one aligned b64 load:
//   Wp2[(kp>>1)*192 + n*2 + (kp&1)] = W[n][kp]  (with per-joint padding)
// ------------------------------------------------------------------
__global__ __launch_bounds__(256)
void pack_w_kernel(const float* __restrict__ reg_w, float* __restrict__ Wp2) {
    int idx = blockIdx.x * 256 + threadIdx.x;
    if (idx >= KPAD * NREG) return;
    int kp = idx / NREG, n = idx % NREG;
    int j = kp / JCP, r = kp % JCP;
    float v = (r < JC) ? reg_w[(size_t)n * DPAD + j * JC + r] : 0.0f;
    Wp2[(size_t)(kp >> 1) * 192 + n * 2 + (kp & 1)] = v;
}

// ------------------------------------------------------------------
// Kernel 4: GEMM  Gpart[chunk][3072,96] = X[:, Kchunk] * W[Kchunk, :]
// One wave per block; two 16-row M tiles (32 rows) x N=96 (12 accums).
// Per K-step: 2 A b64 loads + 6 B b64 loads + 12 f32 WMMAs.
// A: lane<16 holds K=kp,kp+1 of row M=lane; lane>=16 holds K=kp+2,kp+3.
// B: vgpr0={K=kp lanes0-15, K=kp+2 lanes16-31}, vgpr1={kp+1, kp+3}.
// ------------------------------------------------------------------
__global__ __launch_bounds__(32)
void gemm_wmma_kernel(const float* __restrict__ fe1, const float* __restrict__ fmd,
                      const float* __restrict__ fe2,
                      const float* __restrict__ ie1, const float* __restrict__ imd,
                      const float* __restrict__ ie2,
                      const float* __restrict__ Wp2, float* __restrict__ Gpart) {
    const int lane  = threadIdx.x;
    const int hf    = lane >> 4;          // 0/1: which K pair
    const int l15   = lane & 15;
    const int m0    = blockIdx.x * 32;    // 32-row strip, never straddles a stream
    const int chunk = blockIdx.y;         // K chunk: joints [chunk*7, chunk*7+7)
    const int mA    = m0 + l15;           // A row, tile 0
    const int mB    = m0 + 16 + l15;      // A row, tile 1
    const int s     = mA >> 10;           // same stream for both tiles
    const int bA    = mA & 1023;
    const int bB    = mB & 1023;

    const float* fbase = (s == 0) ? fe1 : (s == 1) ? fmd : fe2;
    const float* ibase = (s == 0) ? ie1 : (s == 1) ? imd : ie2;
    const float* frow0 = fbase + (size_t)bA * (NJ * PF);
    const float* frow1 = fbase + (size_t)bB * (NJ * PF);
    const float* irow0 = ibase + (size_t)bA * (NJ * 3);
    const float* irow1 = ibase + (size_t)bB * (NJ * 3);

    v8f c0 = {}, c1 = {}, c2 = {}, c3 = {}, c4 = {}, c5 = {};
    v8f d0 = {}, d1 = {}, d2 = {}, d3 = {}, d4 = {}, d5 = {};

    const int jbeg = chunk * JPC;
    for (int j = jbeg; j < jbeg + JPC; ++j) {
        const float* fj0 = frow0 + j * PF;
        const float* fj1 = frow1 + j * PF;
        const float* ij0 = irow0 + j * 3;
        const float* ij1 = irow1 + j * 3;
        const int    kh0 = (j * JCP) >> 1;      // kp0/2 (kp0 divisible by 4)

        // 128 full-feature K-steps (r0 = 0..508), contiguous b64 loads
#pragma unroll 2
        for (int t = 0; t < 128; ++t) {
            const int r = 4 * t + 2 * hf;
            v2f a0 = *(const v2f*)(fj0 + r);
            v2f a1 = *(const v2f*)(fj1 + r);
            const v2f* wv = (const v2f*)(Wp2 + (size_t)(kh0 + 2 * t + hf) * 192) + l15;
            v2f b0 = wv[0],  b1 = wv[16], b2 = wv[32];
            v2f b3 = wv[48], b4 = wv[64], b5 = wv[80];
            c0 = __builtin_amdgcn_wmma_f32_16x16x4_f32(false, a0, false, b0, (short)0, c0, false, false);
            c1 = __builtin_amdgcn_wmma_f32_16x16x4_f32(false, a0, false, b1, (short)0, c1, false, false);
            c2 = __builtin_amdgcn_wmma_f32_16x16x4_f32(false, a0, false, b2, (short)0, c2, false, false);
            c3 = __builtin_amdgcn_wmma_f32_16x16x4_f32(false, a0, false, b3, (short)0, c3, false, false);
            c4 = __builtin_amdgcn_wmma_f32_16x16x4_f32(false, a0, false, b4, (short)0, c4, false, false);
            c5 = __builtin_amdgcn_wmma_f32_16x16x4_f32(false, a0, false, b5, (short)0, c5, false, false);
            d0 = __builtin_amdgcn_wmma_f32_16x16x4_f32(false, a1, false, b0, (short)0, d0, false, false);
            d1 = __builtin_amdgcn_wmma_f32_16x16x4_f32(false, a1, false, b1, (short)0, d1, false, false);
            d2 = __builtin_amdgcn_wmma_f32_16x16x4_f32(false, a1, false, b2, (short)0, d2, false, false);
            d3 = __builtin_amdgcn_wmma_f32_16x16x4_f32(false, a1, false, b3, (short)0, d3, false, false);
            d4 = __builtin_amdgcn_wmma_f32_16x16x4_f32(false, a1, false, b4, (short)0, d4, false, false);
            d5 = __builtin_amdgcn_wmma_f32_16x16x4_f32(false, a1, false, b5, (short)0, d5, false, false);
        }
        // tail K-step: r0 = 512 -> elements {img0, img1, img2, 0} (scalar, 4B-aligned)
        {
            v2f a0, a1;
            a0.x = ij0[2 * hf]; a0.y = (hf == 0) ? ij0[1] : 0.0f;
            a1.x = ij1[2 * hf]; a1.y = (hf == 0) ? ij1[1] : 0.0f;
            const v2f* wv = (const v2f*)(Wp2 + (size_t)(kh0 + 256 + hf) * 192) + l15;
            v2f b0 = wv[0],  b1 = wv[16], b2 = wv[32];
            v2f b3 = wv[48], b4 = wv[64], b5 = wv[80];
            c0 = __builtin_amdgcn_wmma_f32_16x16x4_f32(false, a0, false, b0, (short)0, c0, false, false);
            c1 = __builtin_amdgcn_wmma_f32_16x16x4_f32(false, a0, false, b1, (short)0, c1, false, false);
            c2 = __builtin_amdgcn_wmma_f32_16x16x4_f32(false, a0, false, b2, (short)0, c2, false, false);
            c3 = __builtin_amdgcn_wmma_f32_16x16x4_f32(false, a0, false, b3, (short)0, c3, false, false);
            c4 = __builtin_amdgcn_wmma_f32_16x16x4_f32(false, a0, false, b4, (short)0, c4, false, false);
            c5 = __builtin_amdgcn_wmma_f32_16x16x4_f32(false, a0, false, b5, (short)0, c5, false, false);
            d0 = __builtin_amdgcn_wmma_f32_16x16x4_f32(false, a1, false, b0, (short)0, d0, false, false);
            d1 = __builtin_amdgcn_wmma_f32_16x16x4_f32(false, a1, false, b1, (short)0, d1, false, false);
            d2 = __builtin_amdgcn_wmma_f32_16x16x4_f32(false, a1, false, b2, (short)0, d2, false, false);
            d3 = __builtin_amdgcn_wmma_f32_16x16x4_f32(false, a1, false, b3, (short)0, d3, false, false);
            d4 = __builtin_amdgcn_wmma_f32_16x16x4_f32(false, a1, false, b4, (short)0, d4, false, false);
            d5 = __builtin_amdgcn_wmma_f32_16x16x4_f32(false, a1, false, b5, (short)0, d5, false, false);
        }
    }

    // store: C/D layout: lanes0-15 -> M=v, lanes16-31 -> M=v+8; N = nt*16 + l15
    float* Gc = Gpart + (size_t)chunk * MTOT * NREG;
    const v8f cc[6] = {c0, c1, c2, c3, c4, c5};
    const v8f dd[6] = {d0, d1, d2, d3, d4, d5};
#pragma unroll
    for (int nt = 0; nt < 6; ++nt)
#pragma unroll
        for (int v = 0; v < 8; ++v) {
            Gc[(size_t)(m0 + 8 * hf + v) * NREG + nt * 16 + l15]      = cc[nt][v];
            Gc[(size_t)(m0 + 16 + 8 * hf + v) * NREG + nt * 16 + l15] = dd[nt][v];
        }
}

// ------------------------------------------------------------------
// Kernel 5: sum 3 K-partials (fixed order => deterministic), then
// ancestor chain (6x6 corrections) + rot6d -> axis-angle.
// One thread per (stream, batch) row.
// ------------------------------------------------------------------
__global__ __launch_bounds__(128)
void chain_kernel(const float* __restrict__ Gpart, const float* __restrict__ reg_w,
                  const float* __restrict__ reg_b, float* __restrict__ out) {
    int m = blockIdx.x * 128 + threadIdx.x;
    if (m >= MTOT) return;
    const int s = m >> 10, b = m & 1023;

    constexpr int ANC[16][3] = {
        {-1,-1,-1},{0,-1,-1},{0,1,-1},{0,1,2},{0,-1,-1},{0,4,-1},{0,4,5},{0,-1,-1},
        {0,7,-1},{0,7,8},{0,-1,-1},{0,10,-1},{0,10,11},{0,-1,-1},{0,13,-1},{0,13,14}};
    constexpr int NANC[16] = {0,1,2,3,1,2,3,1,2,3,1,2,3,1,2,3};

    float p[96];
    const float* g0 = Gpart + (size_t)m * NREG;
    const float* g1 = g0 + (size_t)MTOT * NREG;
    const float* g2 = g1 + (size_t)MTOT * NREG;

#pragma unroll
    for (int i = 0; i < 16; ++i) {
#pragma unroll
        for (int o = 0; o < 6; ++o) {
            const int n = i * 6 + o;
            float acc = (g0[n] + g1[n] + g2[n]) + reg_b[n];
            const float* wr = reg_w + (size_t)n * DPAD + FEATD;
#pragma unroll
            for (int k = 0; k < 3; ++k) {
                if (k < NANC[i]) {
                    const int aj = ANC[i][k];
#pragma unroll
                    for (int c = 0; c < 6; ++c) acc += wr[6 * k + c] * p[aj * 6 + c];
                }
            }
            p[n] = acc;
        }
        // rot6d -> axis-angle
        const float v0 = p[i*6+0], v1 = p[i*6+1], v2 = p[i*6+2];
        const float v3 = p[i*6+3], v4 = p[i*6+4], v5 = p[i*6+5];
        float a1x = v0, a1y = v2, a1z = v4;        // x[...,0]
        float a2x = v1, a2y = v3, a2z = v5;        // x[...,1]
        float n1 = fmaxf(sqrtf(a1x*a1x + a1y*a1y + a1z*a1z), 1e-12f);
        float b1x = a1x / n1, b1y = a1y / n1, b1z = a1z / n1;
        float d = b1x*a2x + b1y*a2y + b1z*a2z;
        float px = a2x - d*b1x, py = a2y - d*b1y, pz = a2z - d*b1z;
        float n2 = fmaxf(sqrtf(px*px + py*py + pz*pz), 1e-12f);
        float b2x = px / n2, b2y = py / n2, b2z = pz / n2;
        float b3x = b1y*b2z - b1z*b2y;
        float b3y = b1z*b2x - b1x*b2z;
        float b3z = b1x*b2y - b1y*b2x;
        float tr = b1x + b2y + b3z;
        float cosv = fminf(fmaxf((tr - 1.0f) * 0.5f, -1.0f + 1e-6f), 1.0f - 1e-6f);
        float ang = acosf(cosv);
        float sc = ang / (2.0f * sinf(ang));
        float ax = (b2z - b3y) * sc;
        float ay = (b3x - b1z) * sc;
        float az = (b1y - b2x) * sc;
        float* po = out + POSE_OFF + (size_t)s * (BATCH * 48) + (size_t)b * 48 + i * 3;
        po[0] = ax; po[1] = ay; po[2] = az;
    }
}

// ------------------------------------------------------------------
extern "C" void kernel_launch(void* const* d_in, const int* in_sizes, int n_in,
                              void* d_out, int out_size, void* d_ws, size_t ws_size,
                              hipStream_t stream) {
    const float* feat_blur = (const float*)d_in[0];
    const float* fe1 = (const float*)d_in[1];
    const float* fmd = (const float*)d_in[2];
    const float* fe2 = (const float*)d_in[3];
    const float* ie1 = (const float*)d_in[4];
    const float* imd = (const float*)d_in[5];
    const float* ie2 = (const float*)d_in[6];
    const float* shape_w = (const float*)d_in[7];
    const float* shape_b = (const float*)d_in[8];
    const float* cam_w   = (const float*)d_in[9];
    const float* cam_b   = (const float*)d_in[10];
    const float* reg_w   = (const float*)d_in[11];
    const float* reg_b   = (const float*)d_in[12];
    float* out = (float*)d_out;

    char* ws = (char*)d_ws;
    float* pooled = (float*)(ws + WS_POOLED);
    float* Wp2    = (float*)(ws + WS_WP);
    float* Gpart  = (float*)(ws + WS_G);

    // 1. mean pool
    pool_kernel<<<(BATCH * 2048) / 256, 256, 0, stream>>>(feat_blur, pooled);
    // 2. shape/cam heads
    shapecam_kernel<<<BATCH, 256, 0, stream>>>(pooled, shape_w, shape_b, cam_w, cam_b, out);
    // 3. pack W (K-pair interleaved, padded)
    pack_w_kernel<<<(KPAD * NREG + 255) / 256, 256, 0, stream>>>(reg_w, Wp2);
    // 4. big WMMA GEMM, K split over 3 joint chunks -> Gpart
    gemm_wmma_kernel<<<dim3(MTOT / 32, KCHUNKS), 32, 0, stream>>>(
        fe1, fmd, fe2, ie1, imd, ie2, Wp2, Gpart);
    // 5. partial-sum + ancestor chain + rot6d->axis-angle
    chain_kernel<<<(MTOT + 127) / 128, 128, 0, stream>>>(Gpart, reg_w, reg_b, out);
}